// ContrastiveLoss3DTo2D_31602369364167
// MI455X (gfx1250) — compile-verified
//
#include <hip/hip_runtime.h>

// Problem constants (match the reference)
#define BSZ 256
#define DIM 1024
#define MARGIN 0.2f

typedef __attribute__((ext_vector_type(2))) float v2f;
typedef __attribute__((ext_vector_type(8))) float v8f;

// ---------------------------------------------------------------------------
// Kernel 1: scores[i,j] = dot(im[j], s[i,j]) via V_WMMA_F32_16X16X4_F32.
// One wave32 per (i, 16-wide j tile). The wave accumulates a full 16x16
// product diag(S_tile * IM_tile^T) over K=DIM in steps of 4 and keeps only the
// diagonal (the 16 wanted dot products). Memory-bound: s streamed once (NT).
// ---------------------------------------------------------------------------
__global__ void __launch_bounds__(256)
scores_wmma_kernel(const float* __restrict__ im,
                   const float* __restrict__ s,
                   float* __restrict__ scores) {
  const int lane = threadIdx.x & 31;
  const int wave = threadIdx.x >> 5;
  const int flat = blockIdx.x * 8 + wave;   // 0 .. BSZ*(BSZ/16)-1 == 4095
  const int i    = flat >> 4;               // caption row index (0..255)
  const int jb   = (flat & 15) << 4;        // image tile base (0,16,...,240)

  // 32-bit A-matrix 16x4 layout: lanes 0-15 hold M=0..15 with K=0,1 in the two
  // VGPRs; lanes 16-31 hold the same M rows with K=2,3.  B (4x16) mirrors it.
  const int m   = lane & 15;                // matrix row (j within tile)
  const int kph = (lane >> 4) << 1;         // 0 or 2: per-lane K phase

  const float* __restrict__ arow = s  + ((size_t)i * BSZ + (jb + m)) * DIM + kph;
  const float* __restrict__ brow = im + (size_t)(jb + m) * DIM + kph;

  v8f c = {};
#pragma unroll 4
  for (int k = 0; k < DIM; k += 4) {
    // s: 256 MB streamed exactly once -> non-temporal (don't thrash 192MB L2)
    v2f a = __builtin_nontemporal_load((const v2f*)(arow + k));
    // im: 1 MB total, reused by 256 row-waves -> regular (L2-resident) load
    v2f b = *(const v2f*)(brow + k);
    // 8 args: (neg_a, A, neg_b, B, c_mod, C, reuse_a, reuse_b)
    c = __builtin_amdgcn_wmma_f32_16x16x4_f32(
        /*neg_a=*/false, a, /*neg_b=*/false, b,
        /*c_mod=*/(short)0, c, /*reuse_a=*/false, /*reuse_b=*/false);
  }

  // Diagonal extraction from the 16x16 f32 C layout:
  //   VGPR r, lanes 0-15  : M=r,   N=lane      -> (r,r)     at lane r
  //   VGPR r, lanes 16-31 : M=8+r, N=lane-16   -> (8+r,8+r) at lane 24+r
  float myval = 0.0f;
  int   myj   = -1;
#pragma unroll
  for (int r = 0; r < 8; ++r) {
    float cr = c[r];
    if (lane == r)      { myval = cr; myj = r; }
    if (lane == 24 + r) { myval = cr; myj = 8 + r; }
  }
  if (myj >= 0) {
    scores[(size_t)i * BSZ + (jb + myj)] = myval;
  }
}

// ---------------------------------------------------------------------------
// Kernel 2: hardest-negative reduction.
//   cost_s[i]  = max_{j!=i} relu(margin + scores[i,j] - scores[i,i])
//   cost_im[j] = max_{i!=j} relu(margin + scores[i,j] - scores[j,j])
//   out = sum(cost_s) + sum(cost_im)
// One block of 256 threads; thread t owns row t and column t (both use the
// same diagonal element scores[t,t]). 256 KB of L2-resident traffic -> noise.
// ---------------------------------------------------------------------------
__global__ void __launch_bounds__(256)
contrastive_reduce_kernel(const float* __restrict__ scores,
                          float* __restrict__ out) {
  __shared__ float partial[BSZ];
  const int t = threadIdx.x;

  const float di = scores[(size_t)t * BSZ + t];
  float row_max = 0.0f;   // relu floor: off-diag values are >= 0, diag is 0
  float col_max = 0.0f;

  for (int u = 0; u < BSZ; ++u) {
    if (u == t) continue;
    const float s_tu = scores[(size_t)t * BSZ + u];  // row t
    const float s_ut = scores[(size_t)u * BSZ + t];  // column t
    row_max = fmaxf(row_max, fmaxf(MARGIN + s_tu - di, 0.0f));
    col_max = fmaxf(col_max, fmaxf(MARGIN + s_ut - di, 0.0f));
  }

  partial[t] = row_max + col_max;
  __syncthreads();

#pragma unroll
  for (int stride = BSZ / 2; stride > 0; stride >>= 1) {
    if (t < stride) partial[t] += partial[t + stride];
    __syncthreads();
  }
  if (t == 0) out[0] = partial[0];
}

// ---------------------------------------------------------------------------
// Host-side launcher (graph-capture safe: only kernel launches on `stream`).
// ---------------------------------------------------------------------------
extern "C" void kernel_launch(void* const* d_in, const int* in_sizes, int n_in,
                              void* d_out, int out_size, void* d_ws, size_t ws_size,
                              hipStream_t stream) {
  (void)in_sizes; (void)n_in; (void)out_size; (void)ws_size;

  const float* im = (const float*)d_in[0];  // [BSZ, DIM]
  const float* s  = (const float*)d_in[1];  // [BSZ, BSZ, DIM]
  float* scores   = (float*)d_ws;           // [BSZ, BSZ] scratch (256 KB)
  float* out      = (float*)d_out;          // scalar

  // BSZ * (BSZ/16) = 4096 waves, 8 waves (256 threads) per block -> 512 blocks
  scores_wmma_kernel<<<512, 256, 0, stream>>>(im, s, scores);
  contrastive_reduce_kernel<<<1, 256, 0, stream>>>(scores, out);
}